// MapLoss_37615323578737
// MI455X (gfx1250) — compile-verified
//
#include <hip/hip_runtime.h>
#include <stdint.h>

#define NTHREADS 256
#define NPIX (512 * 512)
#define NSAMP 32
#define NSLOTS 64            // slot = map*32 + sample; map0=region, map1=affinity
#define K1_CHUNKS 16

#define MODE_SELECT   0u
#define MODE_FALLBACK 1u
#define MODE_SUMALL   2u
#define MODE_ZERO     3u

typedef float v2f __attribute__((ext_vector_type(2)));
typedef float v8f __attribute__((ext_vector_type(8)));

struct WS {
  float* posSumP; float* posCntP; float* negSumP; float* negCntP; // [NSLOTS*K1_CHUNKS]
  float* posLoss; float* result;                                  // [NSLOTS]
  unsigned* mode; unsigned* kval; unsigned* need; unsigned* prefix; // [NSLOTS]
  unsigned* hist;                                                 // [NSLOTS*256]
};

// ---- wave32 sum using V_WMMA_F32_16X16X4_F32 with an all-ones B matrix ----
// A(16x4): lane l<16 holds A[l][0..1], lane l>=16 holds A[l-16][2..3].
// With a = (x, 0): rowsum_m = x_m + x_{m+16};  D[m][n] = rowsum_m for all n.
// Lane n<16 holds D[0..7][n] in c[0..7]; lane n+16 holds D[8..15][n].
// sum(c) on lane n is sum_{m=0..7} rowsum; xor-16 shuffle adds the other half.
__device__ __forceinline__ float wave_sum_wmma(float x) {
  v2f a; a[0] = x;    a[1] = 0.0f;
  v2f b; b[0] = 1.0f; b[1] = 1.0f;
  v8f c = {0.f, 0.f, 0.f, 0.f, 0.f, 0.f, 0.f, 0.f};
  c = __builtin_amdgcn_wmma_f32_16x16x4_f32(false, a, false, b, (short)0, c,
                                            false, false);
  float s = ((c[0] + c[1]) + (c[2] + c[3])) + ((c[4] + c[5]) + (c[6] + c[7]));
  s += __shfl_xor(s, 16);
  return s;
}

__device__ __forceinline__ float block_sum(float x, float* red) {
  float w = wave_sum_wmma(x);
  __syncthreads();
  if ((threadIdx.x & 31) == 0) red[threadIdx.x >> 5] = w;
  __syncthreads();
  float r = 0.f;
  if (threadIdx.x == 0) {
#pragma unroll
    for (int i = 0; i < NTHREADS / 32; ++i) r += red[i];
  }
  return r; // valid on thread 0
}

// =====================================================================
// K1: fused stats pass. Reads all 5 input streams exactly once, staged
// through LDS with double-buffered async-to-LDS loads (ASYNCcnt).
// =====================================================================
__global__ __launch_bounds__(NTHREADS)
void k1_stats(const float* __restrict__ rgt, const float* __restrict__ agt,
              const float* __restrict__ rpd, const float* __restrict__ apd,
              const float* __restrict__ msk, WS ws)
{
  const int samp  = blockIdx.x >> 4;
  const int chunk = blockIdx.x & (K1_CHUNKS - 1);
  const size_t base = (size_t)samp * NPIX + (size_t)chunk * (NPIX / K1_CHUNKS);
  const float4* s4[5];
  s4[0] = (const float4*)(rgt + base);
  s4[1] = (const float4*)(agt + base);
  s4[2] = (const float4*)(rpd + base);
  s4[3] = (const float4*)(apd + base);
  s4[4] = (const float4*)(msk + base);
  const int tid = threadIdx.x;
  constexpr int ITER = (NPIX / K1_CHUNKS) / 4 / NTHREADS; // 16

  __shared__ float4 tile[5][2][NTHREADS]; // 40 KB
  unsigned lds[5][2];
#pragma unroll
  for (int a = 0; a < 5; ++a) {
    lds[a][0] = (unsigned)(uintptr_t)&tile[a][0][tid];
    lds[a][1] = (unsigned)(uintptr_t)&tile[a][1][tid];
  }

  auto issue = [&](int buf, int i) {
#pragma unroll
    for (int a = 0; a < 5; ++a) {
      unsigned long long ga =
          (unsigned long long)(uintptr_t)(s4[a] + (size_t)i * NTHREADS + tid);
      asm volatile("global_load_async_to_lds_b128 %0, %1, off"
                   :: "v"(lds[a][buf]), "v"(ga) : "memory");
    }
  };

  float psR = 0, pcR = 0, nsR = 0, ncR = 0;
  float psA = 0, pcA = 0, nsA = 0, ncA = 0;

  auto one = [&](float G, float P, float Q, float th,
                 float& ps, float& pc, float& ns, float& nc) {
    bool pos = G > th;
    float pp = (pos && (P > 1.0f)) ? 1.0f : P;
    float d = pp - G;
    float pl = d * d * Q;
    if (pos) { ps += pl; pc += 1.0f; } else { ns += pl; nc += 1.0f; }
  };

  issue(0, 0);
  for (int i = 0; i < ITER; ++i) {
    if (i + 1 < ITER) {
      issue((i + 1) & 1, i + 1);
      asm volatile("s_wait_asynccnt 0x5" ::: "memory"); // oldest 5 done
    } else {
      asm volatile("s_wait_asynccnt 0x0" ::: "memory");
    }
    const int b = i & 1;
    float4 g  = tile[0][b][tid];
    float4 a  = tile[1][b][tid];
    float4 pg = tile[2][b][tid];
    float4 pa = tile[3][b][tid];
    float4 mk = tile[4][b][tid];
    one(g.x, pg.x, mk.x, 0.60f, psR, pcR, nsR, ncR);
    one(g.y, pg.y, mk.y, 0.60f, psR, pcR, nsR, ncR);
    one(g.z, pg.z, mk.z, 0.60f, psR, pcR, nsR, ncR);
    one(g.w, pg.w, mk.w, 0.60f, psR, pcR, nsR, ncR);
    one(a.x, pa.x, mk.x, 0.65f, psA, pcA, nsA, ncA);
    one(a.y, pa.y, mk.y, 0.65f, psA, pcA, nsA, ncA);
    one(a.z, pa.z, mk.z, 0.65f, psA, pcA, nsA, ncA);
    one(a.w, pa.w, mk.w, 0.65f, psA, pcA, nsA, ncA);
  }

  __shared__ float red[NTHREADS / 32];
  const int sR = samp, sA = NSAMP + samp;
  float t;
  t = block_sum(psR, red); if (tid == 0) ws.posSumP[sR * K1_CHUNKS + chunk] = t;
  t = block_sum(pcR, red); if (tid == 0) ws.posCntP[sR * K1_CHUNKS + chunk] = t;
  t = block_sum(nsR, red); if (tid == 0) ws.negSumP[sR * K1_CHUNKS + chunk] = t;
  t = block_sum(ncR, red); if (tid == 0) ws.negCntP[sR * K1_CHUNKS + chunk] = t;
  t = block_sum(psA, red); if (tid == 0) ws.posSumP[sA * K1_CHUNKS + chunk] = t;
  t = block_sum(pcA, red); if (tid == 0) ws.posCntP[sA * K1_CHUNKS + chunk] = t;
  t = block_sum(nsA, red); if (tid == 0) ws.negSumP[sA * K1_CHUNKS + chunk] = t;
  t = block_sum(ncA, red); if (tid == 0) ws.negCntP[sA * K1_CHUNKS + chunk] = t;
}

// ---- streaming recompute of pre_loss for the cold select path ----
template <class F>
__device__ __forceinline__ void stream_recompute(const float* __restrict__ g,
                                                 const float* __restrict__ p,
                                                 const float* __restrict__ q,
                                                 float th, F f)
{
  const float4* g4 = (const float4*)g;
  const float4* p4 = (const float4*)p;
  const float4* q4 = (const float4*)q;
  const int tid = threadIdx.x;
  const int ITER = NPIX / 4 / NTHREADS; // 256
  for (int i = 0; i < ITER; ++i) {
    const int idx = i * NTHREADS + tid;
    if (i + 8 < ITER) {
      __builtin_prefetch(g4 + idx + 8 * NTHREADS, 0, 0);
      __builtin_prefetch(p4 + idx + 8 * NTHREADS, 0, 0);
      __builtin_prefetch(q4 + idx + 8 * NTHREADS, 0, 0);
    }
    float4 G = g4[idx], P = p4[idx], Q = q4[idx];
    auto one = [&](float gg, float pp, float qq) {
      bool pos = gg > th;
      float pc = (pos && (pp > 1.0f)) ? 1.0f : pp;
      float d = pc - gg;
      f(d * d * qq, pos);
    };
    one(G.x, P.x, Q.x); one(G.y, P.y, Q.y);
    one(G.z, P.z, Q.z); one(G.w, P.w, Q.w);
  }
}

// =====================================================================
// K2: combine partials, decide mode; SELECT/FALLBACK builds byte-3 hist.
// =====================================================================
__global__ __launch_bounds__(NTHREADS)
void k2_mode_hist(const float* __restrict__ rgt, const float* __restrict__ agt,
                  const float* __restrict__ rpd, const float* __restrict__ apd,
                  const float* __restrict__ msk, WS ws)
{
  const int slot = blockIdx.x;
  const int tid = threadIdx.x;
  float ps = 0, pc = 0, ns = 0, nc = 0;
  for (int c = 0; c < K1_CHUNKS; ++c) {
    ps += ws.posSumP[slot * K1_CHUNKS + c];
    pc += ws.posCntP[slot * K1_CHUNKS + c];
    ns += ws.negSumP[slot * K1_CHUNKS + c];
    nc += ws.negCntP[slot * K1_CHUNKS + c];
  }
  unsigned md, k = 0;
  float posloss = (pc > 0.f) ? ps / pc : 0.f;
  if (pc == 0.f) { md = MODE_FALLBACK; k = 500u; }   // top-500 over all pixels
  else {
    float kf = fminf(3.f * pc, nc);                  // exact: ints < 2^24
    k = (unsigned)kf;
    if (k == 0u) md = MODE_ZERO;
    else if (kf >= nc) md = MODE_SUMALL;             // top-k == all negatives
    else md = MODE_SELECT;
  }
  if (tid == 0) {
    ws.mode[slot] = md; ws.kval[slot] = k; ws.need[slot] = k; ws.prefix[slot] = 0u;
    ws.posLoss[slot] = posloss;
    if (md == MODE_ZERO)        ws.result[slot] = posloss;
    else if (md == MODE_SUMALL) ws.result[slot] = posloss + ns / nc;
  }
  if (md == MODE_ZERO || md == MODE_SUMALL) return;  // uniform per block

  __shared__ unsigned h[256];
  h[tid] = 0u;
  __syncthreads();
  const int m = slot >> 5, s = slot & 31;
  const size_t base = (size_t)s * NPIX;
  stream_recompute((m ? agt : rgt) + base, (m ? apd : rpd) + base, msk + base,
                   m ? 0.65f : 0.60f, [&](float v, bool pos) {
    if (!pos) atomicAdd(&h[__float_as_uint(v) >> 24], 1u);
  });
  __syncthreads();
  ws.hist[slot * 256 + tid] = h[tid];
}

// =====================================================================
// K3/K4/K5: scan previous histogram (top-down), refine prefix, build
// next-byte histogram restricted to the selected prefix.
// =====================================================================
__global__ __launch_bounds__(NTHREADS)
void k3_level(const float* __restrict__ rgt, const float* __restrict__ agt,
              const float* __restrict__ rpd, const float* __restrict__ apd,
              const float* __restrict__ msk, WS ws, int scan_shift)
{
  const int slot = blockIdx.x;
  const int tid = threadIdx.x;
  const unsigned md = ws.mode[slot];
  if (md != MODE_SELECT && md != MODE_FALLBACK) return;
  __shared__ unsigned h[256];
  __shared__ unsigned s_pfx;
  h[tid] = ws.hist[slot * 256 + tid];
  __syncthreads();
  if (tid == 0) {
    unsigned nd = ws.need[slot], above = 0; int sel = 0;
    for (int b = 255; b >= 0; --b) {
      unsigned c = h[b];
      if (above + c >= nd) { sel = b; nd -= above; break; }
      above += c;
    }
    unsigned pfx = ws.prefix[slot] | ((unsigned)sel << scan_shift);
    ws.prefix[slot] = pfx; ws.need[slot] = nd; s_pfx = pfx;
  }
  __syncthreads();
  const unsigned pfx = s_pfx;
  const int ms_ = scan_shift;
  const int ns_ = scan_shift - 8;
  h[tid] = 0u;
  __syncthreads();
  const int m = slot >> 5, s = slot & 31;
  const size_t base = (size_t)s * NPIX;
  stream_recompute((m ? agt : rgt) + base, (m ? apd : rpd) + base, msk + base,
                   m ? 0.65f : 0.60f, [&](float v, bool pos) {
    unsigned bits = __float_as_uint(v);
    if (!pos && ((bits >> ms_) == (pfx >> ms_)))
      atomicAdd(&h[(bits >> ns_) & 255u], 1u);
  });
  __syncthreads();
  ws.hist[slot * 256 + tid] = h[tid];
}

// =====================================================================
// K6: final scan -> exact threshold T and tie count r; sum of values
// strictly above T; top-k sum = sum_gt + r * T.
// =====================================================================
__global__ __launch_bounds__(NTHREADS)
void k6_final(const float* __restrict__ rgt, const float* __restrict__ agt,
              const float* __restrict__ rpd, const float* __restrict__ apd,
              const float* __restrict__ msk, WS ws)
{
  const int slot = blockIdx.x;
  const int tid = threadIdx.x;
  const unsigned md = ws.mode[slot];
  if (md != MODE_SELECT && md != MODE_FALLBACK) return;
  __shared__ unsigned h[256];
  __shared__ unsigned s_T, s_r;
  h[tid] = ws.hist[slot * 256 + tid];
  __syncthreads();
  if (tid == 0) {
    unsigned nd = ws.need[slot], above = 0; int sel = 0;
    for (int b = 255; b >= 0; --b) {
      unsigned c = h[b];
      if (above + c >= nd) { sel = b; nd -= above; break; }
      above += c;
    }
    s_T = ws.prefix[slot] | (unsigned)sel;
    s_r = nd;
  }
  __syncthreads();
  const unsigned T = s_T, r = s_r;
  const int m = slot >> 5, s = slot & 31;
  const size_t base = (size_t)s * NPIX;
  float acc = 0.f;
  stream_recompute((m ? agt : rgt) + base, (m ? apd : rpd) + base, msk + base,
                   m ? 0.65f : 0.60f, [&](float v, bool pos) {
    if (!pos && (__float_as_uint(v) > T)) acc += v;
  });
  __shared__ float red[NTHREADS / 32];
  float tot = block_sum(acc, red);
  if (tid == 0) {
    float mean = (tot + (float)r * __uint_as_float(T)) / (float)ws.kval[slot];
    ws.result[slot] = (md == MODE_FALLBACK) ? mean : (ws.posLoss[slot] + mean);
  }
}

__global__ void k7_out(WS ws, float* out) {
  if (threadIdx.x == 0) {
    float cs = 0.f, af = 0.f;
    for (int i = 0; i < NSAMP; ++i) cs += ws.result[i];
    for (int i = 0; i < NSAMP; ++i) af += ws.result[NSAMP + i];
    out[0] = 2.0f * cs / (float)NSAMP + af / (float)NSAMP;
  }
}

extern "C" void kernel_launch(void* const* d_in, const int* in_sizes, int n_in,
                              void* d_out, int out_size, void* d_ws, size_t ws_size,
                              hipStream_t stream)
{
  const float* rgt = (const float*)d_in[0];
  const float* agt = (const float*)d_in[1];
  const float* rpd = (const float*)d_in[2];
  const float* apd = (const float*)d_in[3];
  const float* msk = (const float*)d_in[4];
  (void)in_sizes; (void)n_in; (void)out_size; (void)ws_size;

  char* w = (char*)d_ws;
  WS ws;
  auto take = [&](size_t bytes) -> char* {
    char* r = w;
    w += (bytes + 255) & ~(size_t)255;
    return r;
  };
  ws.posSumP = (float*)take(NSLOTS * K1_CHUNKS * sizeof(float));
  ws.posCntP = (float*)take(NSLOTS * K1_CHUNKS * sizeof(float));
  ws.negSumP = (float*)take(NSLOTS * K1_CHUNKS * sizeof(float));
  ws.negCntP = (float*)take(NSLOTS * K1_CHUNKS * sizeof(float));
  ws.posLoss = (float*)take(NSLOTS * sizeof(float));
  ws.result  = (float*)take(NSLOTS * sizeof(float));
  ws.mode    = (unsigned*)take(NSLOTS * sizeof(unsigned));
  ws.kval    = (unsigned*)take(NSLOTS * sizeof(unsigned));
  ws.need    = (unsigned*)take(NSLOTS * sizeof(unsigned));
  ws.prefix  = (unsigned*)take(NSLOTS * sizeof(unsigned));
  ws.hist    = (unsigned*)take(NSLOTS * 256 * sizeof(unsigned));

  k1_stats<<<NSAMP * K1_CHUNKS, NTHREADS, 0, stream>>>(rgt, agt, rpd, apd, msk, ws);
  k2_mode_hist<<<NSLOTS, NTHREADS, 0, stream>>>(rgt, agt, rpd, apd, msk, ws);
  k3_level<<<NSLOTS, NTHREADS, 0, stream>>>(rgt, agt, rpd, apd, msk, ws, 24);
  k3_level<<<NSLOTS, NTHREADS, 0, stream>>>(rgt, agt, rpd, apd, msk, ws, 16);
  k3_level<<<NSLOTS, NTHREADS, 0, stream>>>(rgt, agt, rpd, apd, msk, ws, 8);
  k6_final<<<NSLOTS, NTHREADS, 0, stream>>>(rgt, agt, rpd, apd, msk, ws);
  k7_out<<<1, 32, 0, stream>>>(ws, (float*)d_out);
}